// ResonanceFidelityLoss_91328184582373
// MI455X (gfx1250) — compile-verified
//
#include <hip/hip_runtime.h>
#include <math.h>

// ---------------------------------------------------------------------------
// ResonanceFidelityLoss for MI455X (gfx1250, wave32)
//
// Roofline: logits are 16*512*32000 f32 = 1.049 GB -> ~45us @ 23.3 TB/s if
// read exactly once. Everything else (16x4 state evolve) is ~free. So the CE
// kernel is a single-pass streaming online-logsumexp (no second max pass),
// b128 loads, ~1.5 v_exp_f32 per element. The evolve's `cur @ COUPLING`
// (16x4 @ 4x4) maps exactly onto V_WMMA_F32_16X16X4_F32 (A = 16x4 f32),
// run as a 50-step WMMA loop on one wave per state.
// ---------------------------------------------------------------------------

typedef __attribute__((ext_vector_type(2))) float v2f;
typedef __attribute__((ext_vector_type(8))) float v8f;

#define N_B      16
#define N_S      512
#define N_V      32000
#define N_TOK    (N_B * N_S)        // 8192
#define N_V4     (N_V / 4)          // 8000
#define DT_C     0.1f
#define CYCLES_C 50
#define CE_W     0.15f
#define RES_W    0.5f
#define ATT_W    0.2f
#define HAR_W    0.15f

__constant__ float COUP[16] = {
    1.0f, 1.4f, 1.3f, 1.5f,
    0.9f, 1.0f, 0.7f, 1.2f,
    0.6f, 0.8f, 1.0f, 0.5f,
    1.3f, 1.1f, 1.0f, 1.0f };

// ---------------------------------------------------------------------------
// Kernel 1: per-token NLL = logsumexp(row) - row[target]
// One 256-thread block (8 wave32) per token; single pass over 32000 floats.
// ---------------------------------------------------------------------------
__global__ __launch_bounds__(256) void ce_token_kernel(
    const float* __restrict__ logits,
    const int*   __restrict__ targets,
    float*       __restrict__ nll_out)
{
    const int t   = blockIdx.x;          // token id 0..8191
    const int tid = threadIdx.x;
    const float4* __restrict__ row4 =
        (const float4*)(logits + (size_t)t * N_V);

    // Online (m, s) logsumexp state: lse = m + log(s)
    float m = -__builtin_inff();
    float s = 0.0f;

    for (int i = tid; i < N_V4; i += 256) {
        float4 x = row4[i];
        // lookahead ~32KB: emits global_prefetch_b8 (speculative; OOB dropped)
        __builtin_prefetch((const void*)(row4 + i + 2048), 0, 0);

        const float mx = fmaxf(fmaxf(x.x, x.y), fmaxf(x.z, x.w));
        const float ls = __expf(x.x - mx) + __expf(x.y - mx) +
                         __expf(x.z - mx) + __expf(x.w - mx);
        const float M  = fmaxf(m, mx);
        s = s * __expf(m - M) + ls * __expf(mx - M);
        m = M;
    }

    // wave32 reduction of (m, s)
    #pragma unroll
    for (int off = 16; off > 0; off >>= 1) {
        const float om = __shfl_xor(m, off, 32);
        const float os = __shfl_xor(s, off, 32);
        const float M  = fmaxf(m, om);
        s = s * __expf(m - M) + os * __expf(om - M);
        m = M;
    }

    __shared__ float sm[8], ss[8];
    const int wave = tid >> 5;
    const int lane = tid & 31;
    if (lane == 0) { sm[wave] = m; ss[wave] = s; }
    __syncthreads();

    if (tid == 0) {
        float M = sm[0], S = ss[0];
        #pragma unroll
        for (int w = 1; w < 8; ++w) {
            const float om = sm[w], os = ss[w];
            const float nM = fmaxf(M, om);
            S = S * __expf(M - nM) + os * __expf(om - nM);
            M = nM;
        }
        const int   tg = targets[t];
        const float tv = logits[(size_t)t * N_V + tg];
        nll_out[t] = M + __logf(S) - tv;
    }
}

// ---------------------------------------------------------------------------
// Evolve one 16x4 state for 50 steps on a single full wave32 using
// V_WMMA_F32_16X16X4_F32.  EXEC must be all ones in the calling wave.
//
// A layout (ISA 16x4 f32): lane L holds row r = L&15; VGPR0 = K(2*hi),
// VGPR1 = K(2*hi+1) where hi = L>>4.  B layout (4x16): VGPR0 = row 2*hi,
// VGPR1 = row 2*hi+1, column n = L&15 (cols 4..15 zero-padded).
// D layout (16x16 f32 C/D): VGPR v = row v + 8*hi, column n = L&15.
// D -> A remap each step goes through a 64-float LDS tile; the cross-lane
// LDS dependency is enforced with the CDNA5 split counter: s_wait_dscnt 0.
// ---------------------------------------------------------------------------
__device__ void evolve_wave_wmma(const float* __restrict__ g_state,  // 16x4 row-major
                                 float* __restrict__ lds_prod,       // 64 floats scratch
                                 float* __restrict__ lds_out,        // 64 floats result
                                 int lane)
{
    const int r  = lane & 15;
    const int hi = lane >> 4;
    const int n  = lane & 15;

    // Load state into the WMMA A-matrix register layout.
    v2f a;
    a.x = g_state[r * 4 + 2 * hi + 0];
    a.y = g_state[r * 4 + 2 * hi + 1];

    // COUPLING into B layout (loop-invariant), zero-padded to 4x16.
    v2f b;
    b.x = (n < 4) ? COUP[(2 * hi + 0) * 4 + n] : 0.0f;
    b.y = (n < 4) ? COUP[(2 * hi + 1) * 4 + n] : 0.0f;

    for (int it = 0; it < CYCLES_C; ++it) {
        v8f cz = {};
        // D = cur @ COUPLING  (valid columns 0..3)
        v8f d = __builtin_amdgcn_wmma_f32_16x16x4_f32(
            /*neg_a=*/false, a, /*neg_b=*/false, b,
            /*c_mod=*/(short)0, cz, /*reuse_a=*/false, /*reuse_b=*/false);

        // kappa from *current* state: 0.5 + 1/(1 + ||cur_row - 1||)
        float q = (a.x - 1.0f) * (a.x - 1.0f) + (a.y - 1.0f) * (a.y - 1.0f);
        q += __shfl_xor(q, 16, 32);
        const float kappa = 0.5f + 1.0f / (1.0f + sqrtf(q));

        // Scatter valid D columns into a row-major 16x4 LDS tile.
        if (n < 4) {
            #pragma unroll
            for (int v = 0; v < 8; ++v)
                lds_prod[(v + 8 * hi) * 4 + n] = d[v];
        }
        asm volatile("s_wait_dscnt 0" ::: "memory");

        // Gather back into the A layout and apply the clipped Euler step.
        const float p0 = lds_prod[r * 4 + 2 * hi + 0];
        const float p1 = lds_prod[r * 4 + 2 * hi + 1];
        a.x = fminf(fmaxf(a.x + DT_C * kappa * (p0 - a.x), 0.0f), 1.0f);
        a.y = fminf(fmaxf(a.y + DT_C * kappa * (p1 - a.y), 0.0f), 1.0f);
    }

    lds_out[r * 4 + 2 * hi + 0] = a.x;
    lds_out[r * 4 + 2 * hi + 1] = a.y;
    asm volatile("s_wait_dscnt 0" ::: "memory");
}

// ---------------------------------------------------------------------------
// Kernel 2: reduce 8192 NLLs (fixed order -> deterministic), run both WMMA
// evolves (wave 0: src, wave 1: tgt, concurrently), combine all loss terms.
// ---------------------------------------------------------------------------
__global__ __launch_bounds__(256) void finalize_kernel(
    const float* __restrict__ nll,
    const float* __restrict__ src0,
    const float* __restrict__ tgt0,
    float*       __restrict__ out)
{
    __shared__ float red[256];
    __shared__ float prodA[64], prodB[64];
    __shared__ float srcF[64],  tgtF[64];

    const int tid = threadIdx.x;

    // CE sum (deterministic fixed-order tree)
    float acc = 0.0f;
    for (int i = tid; i < N_TOK; i += 256) acc += nll[i];
    red[tid] = acc;
    __syncthreads();
    #pragma unroll
    for (int off = 128; off > 0; off >>= 1) {
        if (tid < off) red[tid] += red[tid + off];
        __syncthreads();
    }

    const int wave = tid >> 5;
    const int lane = tid & 31;
    if (wave == 0)      evolve_wave_wmma(src0, prodA, srcF, lane);
    else if (wave == 1) evolve_wave_wmma(tgt0, prodB, tgtF, lane);
    __syncthreads();

    if (tid == 0) {
        const float ce = red[0] * (1.0f / (float)N_TOK);

        float res = 0.0f, att_s = 0.0f, att_t = 0.0f, har = 0.0f;
        for (int bq = 0; bq < N_B; ++bq) {
            float dd = 0.0f, da = 0.0f, db = 0.0f;
            #pragma unroll
            for (int k = 0; k < 4; ++k) {
                const float xs = srcF[bq * 4 + k];
                const float xt = tgtF[bq * 4 + k];
                dd += (xs - xt) * (xs - xt);
                da += (xs - 1.0f) * (xs - 1.0f);
                db += (xt - 1.0f) * (xt - 1.0f);
            }
            res += sqrtf(dd);
            const float na = sqrtf(da);
            const float nb = sqrtf(db);
            att_s += na;
            att_t += nb;
            har += fabsf(1.0f / (1.0f + na) - 1.0f / (1.0f + nb));
        }
        res   *= (1.0f / (float)N_B);
        att_s *= (1.0f / (float)N_B);
        att_t *= (1.0f / (float)N_B);
        har   *= (1.0f / (float)N_B);

        out[0] = CE_W * ce + RES_W * res +
                 ATT_W * (0.5f * (att_s + att_t)) + HAR_W * har;
    }
}

// ---------------------------------------------------------------------------
extern "C" void kernel_launch(void* const* d_in, const int* in_sizes, int n_in,
                              void* d_out, int out_size, void* d_ws, size_t ws_size,
                              hipStream_t stream) {
    const float* logits  = (const float*)d_in[0];
    const int*   targets = (const int*)d_in[1];
    const float* src     = (const float*)d_in[2];
    const float* tgt     = (const float*)d_in[3];

    float* nll = (float*)d_ws;   // 8192 floats of scratch, rewritten every call

    ce_token_kernel<<<N_TOK, 256, 0, stream>>>(logits, targets, nll);
    finalize_kernel<<<1, 256, 0, stream>>>(nll, src, tgt, (float*)d_out);
}